// VanillaClassifierV2_7292854469121
// MI455X (gfx1250) — compile-verified
//
#include <hip/hip_runtime.h>
#include <stdint.h>

// ---------------------------------------------------------------------------
// Cross-entropy: out = (1/B) * sum_b ( logsumexp(pred[b,:]) - pred[b, target[b]] )
// B = 4096, C = 32000 (f32).  Memory-bound: 524 MB @ 23.3 TB/s => ~22.5 us floor.
// Strategy: 1 workgroup per row; TDM DMA of the full 128 KB row into LDS
// (fits in 320 KB/WGP), exact two-pass softmax from LDS, per-row loss to ws,
// then a single-wave WMMA reduction of the 4096 row losses.
// ---------------------------------------------------------------------------

typedef uint32_t u32x4 __attribute__((ext_vector_type(4)));
typedef int      i32x4 __attribute__((ext_vector_type(4)));
typedef int      i32x8 __attribute__((ext_vector_type(8)));
typedef float    f32x2 __attribute__((ext_vector_type(2)));
typedef float    f32x8 __attribute__((ext_vector_type(8)));

#define ROW_C   32000
#define ROW_C4  (ROW_C / 4)      // 8000 float4
#define BLK     256              // 8 waves (wave32)
#define NWAVES  (BLK / 32)

__device__ __forceinline__ float wave_max(float v) {
#pragma unroll
    for (int off = 16; off > 0; off >>= 1)
        v = fmaxf(v, __shfl_xor(v, off, 32));
    return v;
}
__device__ __forceinline__ float wave_sum(float v) {
#pragma unroll
    for (int off = 16; off > 0; off >>= 1)
        v += __shfl_xor(v, off, 32);
    return v;
}

// ---------------------------------------------------------------------------
// Kernel 1: per-row logsumexp - picked, staged through LDS via the TDM.
// ---------------------------------------------------------------------------
__global__ __launch_bounds__(BLK) void ce_row_kernel(const float* __restrict__ pred,
                                                     const int*   __restrict__ target,
                                                     float*       __restrict__ row_loss) {
    __shared__ float4 smem4[ROW_C4];           // 128 000 bytes: one full row
    __shared__ float  red_max[NWAVES];
    __shared__ float  red_sum[NWAVES];

    const int row = blockIdx.x;
    const int tid = threadIdx.x;
    const int lid = tid & 31;
    const int wid = tid >> 5;

    // ---- TDM: DMA the whole row (HBM -> LDS), issued by wave 0 only ----
    if (wid == 0) {
        const float* rowp = pred + (size_t)row * ROW_C;
        uint64_t ga = (uint64_t)(uintptr_t)rowp;
        // Generic LDS pointer: low 32 bits == LDS byte address (aperture rule).
        uint32_t lds_addr = (uint32_t)(uintptr_t)(void*)&smem4[0];

        u32x4 g0;
        g0[0] = 1u;                                   // count=1 (valid), user mode
        g0[1] = lds_addr;                             // lds_addr
        g0[2] = (uint32_t)ga;                         // global_addr[31:0]
        g0[3] = (uint32_t)((ga >> 32) & 0x1FFFFFFu)   // global_addr[56:32]
              | 0x80000000u;                          // type = 2 ("image")

        i32x8 g1;
        g1[0] = (int)(2u << 16);                      // data_size = 4 bytes
        g1[1] = (int)(((uint32_t)ROW_C & 0xFFFFu) << 16);   // tensor_dim0 lo16
        g1[2] = (int)(((uint32_t)ROW_C >> 16) | (1u << 16)); // dim0 hi16 | tensor_dim1=1 lo16
        g1[3] = (int)((uint32_t)ROW_C << 16);         // dim1 hi16=0 | tile_dim0 = 32000
        g1[4] = 1;                                    // tile_dim1 = 1, tile_dim2 = 0
        g1[5] = ROW_C;                                // tensor_dim0_stride lo32
        g1[6] = 0;                                    // stride hi16 | dim1_stride lo16
        g1[7] = 0;

        i32x4 g2 = {0, 0, 0, 0};
        i32x4 g3 = {0, 0, 0, 0};

#if defined(__clang_major__) && (__clang_major__ >= 23)
        i32x8 g1b = {0, 0, 0, 0, 0, 0, 0, 0};
        __builtin_amdgcn_tensor_load_to_lds(g0, g1, g2, g3, g1b, 0);
#else
        __builtin_amdgcn_tensor_load_to_lds(g0, g1, g2, g3, 0);
#endif
        __builtin_amdgcn_s_wait_tensorcnt(0);         // s_wait_tensorcnt 0
    }
    __syncthreads();                                  // LDS row visible to all waves

    // ---- Pass 1: row max (from LDS) ----
    float m = -INFINITY;
    for (int i = tid; i < ROW_C4; i += BLK) {
        float4 v = smem4[i];
        m = fmaxf(m, fmaxf(fmaxf(v.x, v.y), fmaxf(v.z, v.w)));
    }
    m = wave_max(m);
    if (lid == 0) red_max[wid] = m;
    __syncthreads();
    float M = red_max[0];
#pragma unroll
    for (int k = 1; k < NWAVES; ++k) M = fmaxf(M, red_max[k]);

    // ---- Pass 2: sum of exp (base-2 internally; v_exp_f32 is exp2-native) ----
    const float LOG2E = 1.4426950408889634f;
    const float m2 = M * LOG2E;
    float s = 0.0f;
    for (int i = tid; i < ROW_C4; i += BLK) {
        float4 v = smem4[i];
        s += exp2f(fmaf(v.x, LOG2E, -m2));
        s += exp2f(fmaf(v.y, LOG2E, -m2));
        s += exp2f(fmaf(v.z, LOG2E, -m2));
        s += exp2f(fmaf(v.w, LOG2E, -m2));
    }
    s = wave_sum(s);
    if (lid == 0) red_sum[wid] = s;
    __syncthreads();

    if (tid == 0) {
        float S = red_sum[0];
#pragma unroll
        for (int k = 1; k < NWAVES; ++k) S += red_sum[k];
        const float LN2 = 0.6931471805599453f;
        float lse = M + LN2 * log2f(S);               // natural-log logsumexp
        int t = target[row];
        float xt = ((const float*)smem4)[t];
        row_loss[row] = lse - xt;                     // = -log_softmax picked
    }
}

// ---------------------------------------------------------------------------
// Kernel 2: single-wave WMMA reduction of B row losses (B % 64 == 0).
// A (16x4 f32) carries 64 data values, B = ones  =>  D[i][j] = rowsum_i(A).
// Sum of all 256 D entries = 16 * sum(A); accumulate over chunks, divide by 16.
// Fixed order => deterministic.
// ---------------------------------------------------------------------------
__global__ __launch_bounds__(32) void ce_reduce_wmma_kernel(const float* __restrict__ row_loss,
                                                            float* __restrict__ out,
                                                            int n, float invB) {
    const int lane = threadIdx.x;                     // 32 lanes, EXEC all ones
    f32x8 acc = {};
    f32x2 ones; ones.x = 1.0f; ones.y = 1.0f;

    for (int base = 0; base < n; base += 64) {
        f32x2 a;
        a.x = row_loss[base + lane];
        a.y = row_loss[base + lane + 32];
        // v_wmma_f32_16x16x4_f32 : D = A x ones + C
        acc = __builtin_amdgcn_wmma_f32_16x16x4_f32(
            /*neg_a=*/false, a, /*neg_b=*/false, ones,
            /*c_mod=*/(short)0, acc, /*reuse_a=*/false, /*reuse_b=*/false);
    }

    float t = acc[0] + acc[1] + acc[2] + acc[3] + acc[4] + acc[5] + acc[6] + acc[7];
    t = wave_sum(t);                                  // sum over all lanes
    if (lane == 0) out[0] = t * (1.0f / 16.0f) * invB;
}

// ---------------------------------------------------------------------------
extern "C" void kernel_launch(void* const* d_in, const int* in_sizes, int n_in,
                              void* d_out, int out_size, void* d_ws, size_t ws_size,
                              hipStream_t stream) {
    const float* pred   = (const float*)d_in[0];      // [B, C] f32
    const int*   target = (const int*)d_in[1];        // [B]    i32
    float*       out    = (float*)d_out;              // [1]    f32
    float*       rloss  = (float*)d_ws;               // B floats of scratch

    const int B = in_sizes[1];                        // 4096
    // C is baked to 32000 (LDS row staging); pred layout matches reference.

    ce_row_kernel<<<B, BLK, 0, stream>>>(pred, target, rloss);
    ce_reduce_wmma_kernel<<<1, 32, 0, stream>>>(rloss, out, B, 1.0f / (float)B);
}